// SPUTransformer_86268713107945
// MI455X (gfx1250) — compile-verified
//
#include <hip/hip_runtime.h>

// SPU interval-bound transform: elementwise over (l,u) pairs, f32 in/out.
// 128 MiB streamed once -> ~5.6 us roofline at 23.3 TB/s. NT b128 loads/stores,
// wave32 blocks, 4 float4 (8 pairs) per thread, uniform-branch fast path.
//
// Math notes:
//  - slope = (vu-vl)/(u-l) is only consumed inside `cross`, and cross implies
//    l < 0 < u so u-l > 0 strictly => sign(slope) == sign(vu-vl) exactly
//    (IEEE: -0.0 >= 0 is true either way). Division eliminated.
//  - sigmoid(-x)-1 = -e/(1+e), e = exp(x) in (0,1) for x<0; v_rcp_f32 on
//    1+e in (1,2) is ~1 ulp and avoids the cancellation of 1/(1+e)-1.

typedef float v4f __attribute__((ext_vector_type(4)));

struct PairResult { float lo; float hi; };

__device__ __forceinline__ float spu_eval(float x) {
    float pos_v = __fmaf_rn(x, x, -0.5f);
    float e     = __expf(x);                                   // v_mul + v_exp_f32
    float neg_v = -e * __builtin_amdgcn_rcpf(1.0f + e);        // -e/(1+e)
    return (x >= 0.0f) ? pos_v : neg_v;
}

__device__ __forceinline__ PairResult spu_pair(float l, float u) {
    float vl = spu_eval(l);
    float vu = spu_eval(u);
    float d  = vu - vl;                 // sign(d) == sign(slope) within cross

    bool neg   = (u <= 0.0f);
    bool pos   = (l >= 0.0f);
    bool cross = !(neg || pos);
    bool cpos  = cross && (d >= 0.0f);
    bool cneg  = cross && (d <  0.0f);

    PairResult r;
    r.lo = neg ? vl : (cpos ? -0.5f : 0.0f);
    r.hi = neg ? vu : (cpos ? vu : (cneg ? vl : 0.0f));
    return r;
}

#define SPU_ITEMS 4   // float4s per thread -> 4 outstanding b128 loads/lane
#define SPU_BLOCK 256 // 8 wave32 waves per block
#define SPU_TILE  (SPU_BLOCK * SPU_ITEMS)

__device__ __forceinline__ v4f spu_quad(v4f v) {
    PairResult a = spu_pair(v.x, v.y);
    PairResult b = spu_pair(v.z, v.w);
    v4f r;
    r.x = a.lo; r.y = a.hi;
    r.z = b.lo; r.w = b.hi;
    return r;
}

__global__ __launch_bounds__(SPU_BLOCK) void spu_transform_kernel(
    const v4f* __restrict__ in, v4f* __restrict__ out, int n4) {
    int block_base = (int)blockIdx.x * SPU_TILE;

    if (block_base + SPU_TILE <= n4) {
        // Fast path: whole tile in range, no exec masking.
        int base = block_base + (int)threadIdx.x;
        v4f v[SPU_ITEMS];
#pragma unroll
        for (int k = 0; k < SPU_ITEMS; ++k) {
            v[k] = __builtin_nontemporal_load(&in[base + k * SPU_BLOCK]);
        }
#pragma unroll
        for (int k = 0; k < SPU_ITEMS; ++k) {
            __builtin_nontemporal_store(spu_quad(v[k]), &out[base + k * SPU_BLOCK]);
        }
    } else {
        // Tail block (not taken when n4 % SPU_TILE == 0, as here).
#pragma unroll
        for (int k = 0; k < SPU_ITEMS; ++k) {
            int i = block_base + (int)threadIdx.x + k * SPU_BLOCK;
            if (i < n4) {
                v4f v = __builtin_nontemporal_load(&in[i]);
                __builtin_nontemporal_store(spu_quad(v), &out[i]);
            }
        }
    }
}

extern "C" void kernel_launch(void* const* d_in, const int* in_sizes, int n_in,
                              void* d_out, int out_size, void* d_ws, size_t ws_size,
                              hipStream_t stream) {
    (void)n_in; (void)out_size; (void)d_ws; (void)ws_size;

    const v4f* in = (const v4f*)d_in[0];   // bounds: (N,2) f32, interleaved l,u
    v4f* out      = (v4f*)d_out;           // (N,2) f32, interleaved lower,upper

    int total_floats = in_sizes[0];        // 2*N = 16,777,216
    int n4 = total_floats / 4;             // 4,194,304 float4s

    int grid = (n4 + SPU_TILE - 1) / SPU_TILE;   // 4096 blocks, exact

    spu_transform_kernel<<<grid, SPU_BLOCK, 0, stream>>>(in, out, n4);
}